// GNN_84490596647052
// MI455X (gfx1250) — compile-verified
//
#include <hip/hip_runtime.h>

#define DIN   128
#define SLOPE 0.01f

typedef float v2f __attribute__((ext_vector_type(2)));
typedef float v8f __attribute__((ext_vector_type(8)));

// ---------------------------------------------------------------- utilities
__global__ void k_fill(float* __restrict__ p, size_t n, float v) {
  size_t i = (size_t)blockIdx.x * blockDim.x + threadIdx.x;
  if (i < n) p[i] = v;
}

__global__ void k_count_deg(const long long* __restrict__ dst,
                            float* __restrict__ deg, int E) {
  int e = blockIdx.x * blockDim.x + threadIdx.x;
  if (e < E) atomicAdd(&deg[(int)dst[e]], 1.0f);
}

__global__ void k_rsqrt_inplace(float* __restrict__ deg, int n) {
  int i = blockIdx.x * blockDim.x + threadIdx.x;
  if (i < n) {
    float d = deg[i];
    deg[i] = d > 0.0f ? rsqrtf(d) : 0.0f;  // deg >= 1 always (self loop)
  }
}

// ---------------------------------------------------------------- WMMA GEMM
// H[N, DOUT] = X[N, 128] @ W[DOUT, 128]^T    (fp32, V_WMMA_F32_16X16X4_F32)
// Block: 256 threads = 8 waves; each wave computes a 16-row stripe.
template <int DOUT>
__global__ __launch_bounds__(256) void k_gemm_wmma(
    const float* __restrict__ X, const float* __restrict__ W,
    float* __restrict__ H, int Nn) {
  __shared__ float sW[DOUT * DIN];
  for (int i = threadIdx.x; i < DOUT * DIN; i += 256) sW[i] = W[i];
  __syncthreads();

  const int lane = threadIdx.x & 31;
  const int wv   = threadIdx.x >> 5;
  const int m    = lane & 15;   // A row within tile / B,D column within tile
  const int c    = lane >> 4;   // K-half selector (and D row-half selector)

  const int rowBase = blockIdx.x * 128 + wv * 16;
  const int row     = rowBase + m;
  const bool rowOK  = row < Nn;
  const float* xrow = X + (size_t)(rowOK ? row : 0) * DIN;

  // A fragments for K = 0..127: lane L, vgpr r holds A[m, 4s + 2c + r]
  v2f a[32];
#pragma unroll
  for (int s = 0; s < 32; ++s) {
    v2f t = *(const v2f*)(xrow + 4 * s + 2 * c);
    a[s].x = rowOK ? t.x : 0.0f;
    a[s].y = rowOK ? t.y : 0.0f;
  }

#pragma unroll
  for (int ot = 0; ot < DOUT / 16; ++ot) {
    const float* wrow = sW + (size_t)(ot * 16 + m) * DIN;  // B column = W row
    v8f acc = {0.f, 0.f, 0.f, 0.f, 0.f, 0.f, 0.f, 0.f};
#pragma unroll
    for (int s = 0; s < 32; ++s) {
      v2f b = *(const v2f*)(wrow + 4 * s + 2 * c);  // B[4s+2c+r, m]
      acc = __builtin_amdgcn_wmma_f32_16x16x4_f32(
          false, a[s], false, b, (short)0, acc, false, false);
    }
    // D: vgpr r, lane L -> H[rowBase + r + 8c, ot*16 + m]
#pragma unroll
    for (int r = 0; r < 8; ++r) {
      int rr = rowBase + 8 * c + r;
      if (rr < Nn) H[(size_t)rr * DOUT + ot * 16 + m] = acc[r];
    }
  }
}

// ---------------------------------------------------------------- scatter-add
// One wave per edge; each lane owns D/32 contiguous floats (coalesced 512B/wave).
template <int D>
__global__ __launch_bounds__(256) void k_aggregate(
    const long long* __restrict__ src, const long long* __restrict__ dst,
    const float* __restrict__ dinv, const float* __restrict__ H,
    float* __restrict__ agg, int Ne) {
  constexpr int V = D / 32;
  const int lane = threadIdx.x & 31;
  const int wid  = blockIdx.x * 8 + (threadIdx.x >> 5);
  if (wid >= Ne) return;

  const int s = (int)src[wid];
  const int d = (int)dst[wid];
  const float nrm = dinv[s] * dinv[d];

  const float* hs = H + (size_t)s * D + lane * V;
  float*       ad = agg + (size_t)d * D + lane * V;

  if (V == 4) {
    float4 h = *(const float4*)hs;
    atomicAdd(ad + 0, h.x * nrm);
    atomicAdd(ad + 1, h.y * nrm);
    atomicAdd(ad + 2, h.z * nrm);
    atomicAdd(ad + 3, h.w * nrm);
  } else {
    float2 h = *(const float2*)hs;
    atomicAdd(ad + 0, h.x * nrm);
    atomicAdd(ad + 1, h.y * nrm);
  }
}

// ---------------------------------------------------------------- epilogue
// In place on `out` (which accumulated the edge messages):
//   out = [lrelu]( out + dinv[n]^2 * Hpre + b[j] )
template <int D, bool LRELU>
__global__ void k_finalize(float* __restrict__ out,
                           const float* __restrict__ Hpre,
                           const float* __restrict__ dinv,
                           const float* __restrict__ b, size_t total) {
  size_t i = (size_t)blockIdx.x * blockDim.x + threadIdx.x;
  if (i >= total) return;
  int node = (int)(i / D);
  int j    = (int)(i % D);
  float di = dinv[node];
  float v  = out[i] + di * di * Hpre[i] + b[j];
  if (LRELU) v = v > 0.0f ? v : SLOPE * v;
  out[i] = v;
}

// ---------------------------------------------------------------- launcher
extern "C" void kernel_launch(void* const* d_in, const int* in_sizes, int n_in,
                              void* d_out, int out_size, void* d_ws,
                              size_t ws_size, hipStream_t stream) {
  (void)n_in; (void)out_size; (void)ws_size;

  const float*     x    = (const float*)d_in[0];
  const long long* eidx = (const long long*)d_in[1];
  const float*     W1   = (const float*)d_in[2];
  const float*     b1   = (const float*)d_in[3];
  const float*     W2   = (const float*)d_in[4];
  const float*     b2   = (const float*)d_in[5];

  const int N = in_sizes[0] / DIN;
  const int E = in_sizes[1] / 2;
  const long long* src = eidx;
  const long long* dst = eidx + E;

  // workspace: dinv[N] | Hpre[N*128] (layer 2 reuses first N*64 of Hpre)
  float* dinv = (float*)d_ws;
  float* Hpre = dinv + N;

  // output: h1[N*128] | h2[N*64]; each doubles as its own agg buffer
  float* h1 = (float*)d_out;
  float* h2 = h1 + (size_t)N * 128;

  const size_t totOut = (size_t)N * 192;
  const size_t tot1   = (size_t)N * 128;
  const size_t tot2   = (size_t)N * 64;

  // degree -> dinv (self loops give deg an initial 1)
  k_fill<<<(N + 255) / 256, 256, 0, stream>>>(dinv, (size_t)N, 1.0f);
  k_count_deg<<<(E + 255) / 256, 256, 0, stream>>>(dst, dinv, E);
  k_rsqrt_inplace<<<(N + 255) / 256, 256, 0, stream>>>(dinv, N);

  // zero both aggregation regions (h1 + h2)
  k_fill<<<(unsigned)((totOut + 255) / 256), 256, 0, stream>>>(h1, totOut, 0.0f);

  // layer 1
  k_gemm_wmma<128><<<(N + 127) / 128, 256, 0, stream>>>(x, W1, Hpre, N);
  k_aggregate<128><<<(E + 7) / 8, 256, 0, stream>>>(src, dst, dinv, Hpre, h1, E);
  k_finalize<128, true><<<(unsigned)((tot1 + 255) / 256), 256, 0, stream>>>(
      h1, Hpre, dinv, b1, tot1);

  // layer 2 (h1 rows are 128 floats -> valid GEMM input)
  k_gemm_wmma<64><<<(N + 127) / 128, 256, 0, stream>>>(h1, W2, Hpre, N);
  k_aggregate<64><<<(E + 7) / 8, 256, 0, stream>>>(src, dst, dinv, Hpre, h2, E);
  k_finalize<64, false><<<(unsigned)((tot2 + 255) / 256), 256, 0, stream>>>(
      h2, Hpre, dinv, b2, tot2);
}